// LakerPreconditioner_42941083025730
// MI455X (gfx1250) — compile-verified
//
#include <hip/hip_runtime.h>
#include <math.h>

// ---------------------------------------------------------------------------
// LakerPreconditioner on gfx1250 (MI455X): GEMM-only reformulation.
//   inv(Sigma)      -> Newton-Schulz  X <- 2X - X*(Sigma*X)   (warm-started)
//   Sigma^{-1/2}    -> coupled Newton-Schulz sqrt on A = Sigma/n
// Every O(n^3) op is a batched f32 GEMM lowered to v_wmma_f32_16x16x4_f32.
// LDS tiles are stored as (k-even,k-odd) float2 pairs so each WMMA fragment
// is a single aligned ds_load_b64 into an even VGPR pair (no repack movs);
// pair-rows are padded by 128B so the two half-wave K-rows hit disjoint
// LDS bank halves.
// ---------------------------------------------------------------------------

typedef float v2f __attribute__((ext_vector_type(2)));
typedef float v8f __attribute__((ext_vector_type(8)));

#define N_DIM 512
#define NR 64
#define BH 16
#define CCCP_ITERS 10
#define NSINV_ITERS 6
#define NSQRT_ITERS 20
#define GAMMA_C 0.1f
#define RHO_C 0.01f
#define EPS_SG 1e-8f

// GEMM tiling: block = 256 threads = 8 waves (2 along M x 4 along N);
// block tile 128x128, wave tile 64x32 (4x2 WMMA 16x16 tiles), K chunk 16.
#define BM 128
#define BN 128
#define BK 16
#define APITCH (BM + 16)          // v2f pitch: +128B so k2-rows alternate bank halves
#define BPITCH (BN + 16)

__global__ __launch_bounds__(256)
void gemm_wmma_f32(const float* __restrict__ A, const float* __restrict__ B,
                   const float* __restrict__ Cadd, float* __restrict__ C,
                   int M, int N, int K,
                   int lda, int ldb, int ldcadd, int ldc,
                   long strideA, long strideB, long strideCadd, long strideC,
                   int transA, int transB,
                   float alpha, float beta, float delta,
                   const float* __restrict__ betap)
{
    __shared__ v2f As2[BK / 2][APITCH];   // As2[k2][m] = {A[m][2k2], A[m][2k2+1]}
    __shared__ v2f Bs2[BK / 2][BPITCH];   // Bs2[k2][n] = {B[2k2][n], B[2k2+1][n]}
    float* Asf = (float*)&As2[0][0];
    float* Bsf = (float*)&Bs2[0][0];

    if (betap) beta = betap[0];   // device-resident scalar (lambda_reg)

    const int p = blockIdx.z;
    A    += (long)p * strideA;
    B    += (long)p * strideB;
    Cadd += (long)p * strideCadd;
    C    += (long)p * strideC;

    const int rowBase = blockIdx.y * BM;
    const int colBase = blockIdx.x * BN;

    const int t    = threadIdx.x;
    const int lane = t & 31;
    const int wave = t >> 5;    // 0..7
    const int wm   = wave >> 2; // 0..1  (64-row slabs)
    const int wn   = wave & 3;  // 0..3  (32-col slabs)

    v8f acc[4][2];
    #pragma unroll
    for (int i = 0; i < 4; ++i)
        #pragma unroll
        for (int j = 0; j < 2; ++j)
            #pragma unroll
            for (int e = 0; e < 8; ++e) acc[i][j][e] = 0.0f;

    const int half = lane >> 4;     // which K sub-pair this lane consumes
    const int ml   = lane & 15;

    for (int k0 = 0; k0 < K; k0 += BK) {
        // ---- stage A tile (BM x BK) as k-pairs ----
        if (!transA) {
            for (int e = t; e < BM * BK; e += 256) {
                int m = e >> 4, k = e & 15;
                int gr = rowBase + m, gk = k0 + k;
                float v = (gr < M && gk < K) ? A[(long)gr * lda + gk] : 0.0f;
                Asf[(k >> 1) * (APITCH * 2) + m * 2 + (k & 1)] = v;
            }
        } else {
            for (int e = t; e < BM * BK; e += 256) {
                int k = e >> 7, m = e & 127;
                int gr = rowBase + m, gk = k0 + k;
                float v = (gr < M && gk < K) ? A[(long)gk * lda + gr] : 0.0f;
                Asf[(k >> 1) * (APITCH * 2) + m * 2 + (k & 1)] = v;
            }
        }
        // ---- stage B tile (BK x BN) as k-pairs ----
        if (!transB) {
            for (int e = t; e < BN * BK; e += 256) {
                int k = e >> 7, n = e & 127;
                int gc = colBase + n, gk = k0 + k;
                float v = (gc < N && gk < K) ? B[(long)gk * ldb + gc] : 0.0f;
                Bsf[(k >> 1) * (BPITCH * 2) + n * 2 + (k & 1)] = v;
            }
        } else {
            for (int e = t; e < BN * BK; e += 256) {
                int n = e >> 4, k = e & 15;
                int gc = colBase + n, gk = k0 + k;
                float v = (gc < N && gk < K) ? B[(long)gc * ldb + gk] : 0.0f;
                Bsf[(k >> 1) * (BPITCH * 2) + n * 2 + (k & 1)] = v;
            }
        }
        // speculative prefetch of the next K tile (global_prefetch_b8)
        if (k0 + BK < K) {
            int pm = t & 127;
            if (!transA) {
                if (rowBase + pm < M)
                    __builtin_prefetch(A + (long)(rowBase + pm) * lda + (k0 + BK), 0, 1);
            } else {
                __builtin_prefetch(A + (long)(k0 + BK) * lda + rowBase + pm, 0, 1);
            }
            if (!transB) {
                __builtin_prefetch(B + (long)(k0 + BK) * ldb + colBase + pm, 0, 1);
            } else {
                if (colBase + pm < N)
                    __builtin_prefetch(B + (long)(colBase + pm) * ldb + (k0 + BK), 0, 1);
            }
        }
        __syncthreads();

        // CDNA5 f32 WMMA fragment layout:
        //   A 16x4 : lane = M + 16*(K/2), vgpr = K&1  -> one v2f (ds_load_b64)
        //   B 4x16 : lane = N + 16*(K/2), vgpr = K&1  -> one v2f (ds_load_b64)
        #pragma unroll
        for (int kk = 0; kk < BK; kk += 4) {
            const int k2 = (kk >> 1) + half;
            v2f afrag[4];
            #pragma unroll
            for (int ti = 0; ti < 4; ++ti)
                afrag[ti] = As2[k2][wm * 64 + ti * 16 + ml];
            v2f bfrag[2];
            #pragma unroll
            for (int tj = 0; tj < 2; ++tj)
                bfrag[tj] = Bs2[k2][wn * 32 + tj * 16 + ml];
            #pragma unroll
            for (int ti = 0; ti < 4; ++ti)
                #pragma unroll
                for (int tj = 0; tj < 2; ++tj)
                    acc[ti][tj] = __builtin_amdgcn_wmma_f32_16x16x4_f32(
                        false, afrag[ti], false, bfrag[tj], (short)0,
                        acc[ti][tj], false, false);
        }
        __syncthreads();
    }

    // ---- epilogue: D lane layout: row M = vgpr + 8*(lane>=16), col N = lane&15
    #pragma unroll
    for (int ti = 0; ti < 4; ++ti)
        #pragma unroll
        for (int tj = 0; tj < 2; ++tj)
            #pragma unroll
            for (int vg = 0; vg < 8; ++vg) {
                int gr = rowBase + wm * 64 + ti * 16 + vg + 8 * half;
                int gc = colBase + wn * 32 + tj * 16 + ml;
                if (gr < M && gc < N) {
                    float v = alpha * acc[ti][tj][vg];
                    if (beta != 0.0f) v += beta * Cadd[(long)gr * ldcadd + gc];
                    if (gr == gc) v += delta;
                    C[(long)gr * ldc + gc] = v;
                }
            }
}

// ---------------------------------------------------------------------------

__global__ __launch_bounds__(256)
void set_identity_batched(float* __restrict__ Mat)
{
    long idx = (long)blockIdx.x * 256 + threadIdx.x;
    const long total = (long)BH * N_DIM * N_DIM;
    if (idx < total) {
        int ij = (int)(idx % ((long)N_DIM * N_DIM));
        Mat[idx] = ((ij / N_DIM) == (ij % N_DIM)) ? 1.0f : 0.0f;
    }
}

__global__ __launch_bounds__(256)
void scale_copy(float* __restrict__ dst, const float* __restrict__ src, float s)
{
    long idx = (long)blockIdx.x * 256 + threadIdx.x;
    const long total = (long)BH * N_DIM * N_DIM;
    if (idx < total) dst[idx] = s * src[idx];
}

// row-wise L2 normalization of (BH*NR) rows of length N_DIM, in place
__global__ __launch_bounds__(256)
void normalize_rows(float* __restrict__ U)
{
    __shared__ float red[256];
    float* u = U + (long)blockIdx.x * N_DIM;
    float s = 0.0f;
    for (int i = threadIdx.x; i < N_DIM; i += 256) { float v = u[i]; s += v * v; }
    red[threadIdx.x] = s;
    __syncthreads();
    for (int off = 128; off > 0; off >>= 1) {
        if (threadIdx.x < off) red[threadIdx.x] += red[threadIdx.x + off];
        __syncthreads();
    }
    const float inv = 1.0f / (sqrtf(red[0]) + EPS_SG);
    for (int i = threadIdx.x; i < N_DIM; i += 256) u[i] *= inv;
}

// w = ubar / (dot(ubar, Su) + eps), per row
__global__ __launch_bounds__(256)
void compute_w(const float* __restrict__ U, const float* __restrict__ Su,
               float* __restrict__ W)
{
    __shared__ float red[256];
    const float* u = U  + (long)blockIdx.x * N_DIM;
    const float* s = Su + (long)blockIdx.x * N_DIM;
    float*       w = W  + (long)blockIdx.x * N_DIM;
    float d = 0.0f;
    for (int i = threadIdx.x; i < N_DIM; i += 256) d += u[i] * s[i];
    red[threadIdx.x] = d;
    __syncthreads();
    for (int off = 128; off > 0; off >>= 1) {
        if (threadIdx.x < off) red[threadIdx.x] += red[threadIdx.x + off];
        __syncthreads();
    }
    const float inv = 1.0f / (red[0] + EPS_SG);
    for (int i = threadIdx.x; i < N_DIM; i += 256) w[i] = u[i] * inv;
}

__global__ __launch_bounds__(256)
void compute_traces(const float* __restrict__ St, float* __restrict__ traces)
{
    __shared__ float red[256];
    const float* Sp = St + (long)blockIdx.x * N_DIM * N_DIM;
    float s = 0.0f;
    for (int i = threadIdx.x; i < N_DIM; i += 256) s += Sp[(long)i * N_DIM + i];
    red[threadIdx.x] = s;
    __syncthreads();
    for (int off = 128; off > 0; off >>= 1) {
        if (threadIdx.x < off) red[threadIdx.x] += red[threadIdx.x + off];
        __syncthreads();
    }
    if (threadIdx.x == 0) traces[blockIdx.x] = red[0];
}

__global__ __launch_bounds__(256)
void trace_scale(const float* __restrict__ St, const float* __restrict__ traces,
                 float* __restrict__ Sigma)
{
    long idx = (long)blockIdx.x * 256 + threadIdx.x;
    const long total = (long)BH * N_DIM * N_DIM;
    if (idx < total) {
        int p = (int)(idx / ((long)N_DIM * N_DIM));
        Sigma[idx] = St[idx] * ((float)N_DIM / (traces[p] + EPS_SG));
    }
}

// ---------------------------------------------------------------------------

extern "C" void kernel_launch(void* const* d_in, const int* in_sizes, int n_in,
                              void* d_out, int out_size, void* d_ws, size_t ws_size,
                              hipStream_t stream)
{
    (void)in_sizes; (void)n_in; (void)out_size; (void)ws_size;

    const float* kern = (const float*)d_in[0];  // (2,8,512,512)
    const float* lamp = (const float*)d_in[1];  // (1,)
    const float* z    = (const float*)d_in[2];  // (64,2,8,512)
    float* out = (float*)d_out;                 // (2,8,512,512)

    const long MS  = (long)N_DIM * N_DIM;       // per-head matrix size
    const long RS  = (long)NR * N_DIM;          // per-head (64 x 512) block

    // workspace layout (floats)
    float* ubar   = (float*)d_ws;               // BH*NR*N_DIM
    float* su     = ubar + (long)BH * RS;
    float* w      = su   + (long)BH * RS;
    float* traces = w    + (long)BH * RS;       // BH (+pad)
    float* Sigma  = traces + 64;
    float* St     = Sigma + (long)BH * MS;
    float* Tmp    = St    + (long)BH * MS;
    float* X0     = Tmp   + (long)BH * MS;
    float* X1     = X0    + (long)BH * MS;

    const dim3 blk(256);
    const int EW_BLOCKS = (int)(((long)BH * MS + 255) / 256);

    auto launch_gemm = [&](const float* A, const float* B, const float* Cadd,
                           float* C, int M, int N, int K,
                           int lda, int ldb, int ldcadd, int ldc,
                           long sA, long sB, long sCadd, long sC,
                           int tA, int tB, float alpha, float beta, float delta,
                           const float* betap) {
        dim3 grid((N + BN - 1) / BN, (M + BM - 1) / BM, BH);
        hipLaunchKernelGGL(gemm_wmma_f32, grid, blk, 0, stream,
                           A, B, Cadd, C, M, N, K, lda, ldb, ldcadd, ldc,
                           sA, sB, sCadd, sC, tA, tB, alpha, beta, delta, betap);
    };

    // ---- 1) U = Z * K^T + lambda * Z  (per head), then row-normalize -> ubar
    // z[k,p,m]: base p*512, row(k) stride 8192.  kern: row-major 512x512 per p.
    launch_gemm(z, kern, z, ubar,
                NR, N_DIM, N_DIM,
                BH * N_DIM, N_DIM, BH * N_DIM, N_DIM,
                (long)N_DIM, MS, (long)N_DIM, RS,
                0, 1, 1.0f, 0.0f, 0.0f, lamp);
    hipLaunchKernelGGL(normalize_rows, dim3(BH * NR), blk, 0, stream, ubar);

    // ---- 2) Sigma = I, X = I (exact inverse of Sigma0)
    hipLaunchKernelGGL(set_identity_batched, dim3(EW_BLOCKS), blk, 0, stream, Sigma);
    hipLaunchKernelGGL(set_identity_batched, dim3(EW_BLOCKS), blk, 0, stream, X0);

    const float denomG = 1.0f + GAMMA_C / (float)N_DIM;
    const float s1 = (1.0f - RHO_C) * ((float)N_DIM / (float)NR) / denomG;
    const float s2 = (1.0f - RHO_C) * GAMMA_C / denomG + RHO_C;

    float* X  = X0;
    float* Xo = X1;

    for (int t = 0; t < CCCP_ITERS; ++t) {
        // ---- Newton-Schulz refresh of X ~= Sigma^{-1} (warm started)
        if (t > 0) {
            for (int j = 0; j < NSINV_ITERS; ++j) {
                // Tmp = Sigma * X
                launch_gemm(Sigma, X, Tmp, Tmp, N_DIM, N_DIM, N_DIM,
                            N_DIM, N_DIM, N_DIM, N_DIM, MS, MS, MS, MS,
                            0, 0, 1.0f, 0.0f, 0.0f, nullptr);
                // Xo = 2*X - X*Tmp
                launch_gemm(X, Tmp, X, Xo, N_DIM, N_DIM, N_DIM,
                            N_DIM, N_DIM, N_DIM, N_DIM, MS, MS, MS, MS,
                            0, 0, -1.0f, 2.0f, 0.0f, nullptr);
                float* sw = X; X = Xo; Xo = sw;
            }
        }

        // ---- Su = Ubar * X   (X symmetric => equals Sigma^{-1} ubar)
        launch_gemm(ubar, X, su, su, NR, N_DIM, N_DIM,
                    N_DIM, N_DIM, N_DIM, N_DIM, RS, MS, RS, RS,
                    0, 0, 1.0f, 0.0f, 0.0f, nullptr);

        // ---- w = ubar / (sum ubar*Su + eps)
        hipLaunchKernelGGL(compute_w, dim3(BH * NR), blk, 0, stream, ubar, su, w);

        // ---- St = s1 * (W^T Ubar) + s2 * I   (Sigma_tilde, pre-trace-norm)
        launch_gemm(w, ubar, St, St, N_DIM, N_DIM, NR,
                    N_DIM, N_DIM, N_DIM, N_DIM, RS, RS, MS, MS,
                    1, 0, s1, 0.0f, s2, nullptr);

        // ---- Sigma = St * n / (trace(St) + eps)
        hipLaunchKernelGGL(compute_traces, dim3(BH), blk, 0, stream, St, traces);
        hipLaunchKernelGGL(trace_scale, dim3(EW_BLOCKS), blk, 0, stream,
                           St, traces, Sigma);
    }

    // ---- 3) P = Sigma^{-1/2} via coupled Newton-Schulz on A = Sigma/n
    //      Y0 = Sigma/n, Z0 = I;  T = 1.5I - 0.5*Z*Y;  Y<-Y*T;  Z<-T*Z
    //      trace(Sigma)=n  =>  eig(A) in (0,1]  => convergence guaranteed.
    hipLaunchKernelGGL(scale_copy, dim3(EW_BLOCKS), blk, 0, stream,
                       X0, Sigma, 1.0f / (float)N_DIM);
    hipLaunchKernelGGL(set_identity_batched, dim3(EW_BLOCKS), blk, 0, stream, St);

    float* Y  = X0;  float* Yo = X1;
    float* Z  = St;  float* Zo = Sigma;   // Sigma is free after Y0 init
    const float invsq = 1.0f / sqrtf((float)N_DIM);

    for (int it = 0; it < NSQRT_ITERS; ++it) {
        const bool last = (it == NSQRT_ITERS - 1);
        // Tmp = 1.5*I - 0.5 * Z*Y
        launch_gemm(Z, Y, Tmp, Tmp, N_DIM, N_DIM, N_DIM,
                    N_DIM, N_DIM, N_DIM, N_DIM, MS, MS, MS, MS,
                    0, 0, -0.5f, 0.0f, 1.5f, nullptr);
        if (!last) {
            // Yo = Y * Tmp ; Zo = Tmp * Z
            launch_gemm(Y, Tmp, Yo, Yo, N_DIM, N_DIM, N_DIM,
                        N_DIM, N_DIM, N_DIM, N_DIM, MS, MS, MS, MS,
                        0, 0, 1.0f, 0.0f, 0.0f, nullptr);
            launch_gemm(Tmp, Z, Zo, Zo, N_DIM, N_DIM, N_DIM,
                        N_DIM, N_DIM, N_DIM, N_DIM, MS, MS, MS, MS,
                        0, 0, 1.0f, 0.0f, 0.0f, nullptr);
            float* sw;
            sw = Y; Y = Yo; Yo = sw;
            sw = Z; Z = Zo; Zo = sw;
        } else {
            // P = (Tmp * Z) / sqrt(n)  -> directly into d_out
            launch_gemm(Tmp, Z, out, out, N_DIM, N_DIM, N_DIM,
                        N_DIM, N_DIM, N_DIM, N_DIM, MS, MS, MS, MS,
                        0, 0, invsq, 0.0f, 0.0f, nullptr);
        }
    }
}